// FocalHungarianMatcher_36266703847657
// MI455X (gfx1250) — compile-verified
//
#include <hip/hip_runtime.h>
#include <hip/hip_bf16.h>

typedef float v2f __attribute__((ext_vector_type(2)));
typedef float v8f __attribute__((ext_vector_type(8)));
typedef unsigned int u32x4 __attribute__((ext_vector_type(4)));
typedef unsigned int u32x8 __attribute__((ext_vector_type(8)));

#define BS 32
#define NQ 300
#define NC 91
#define NT 48
#define QPAD 304            // 19 tiles of 16
#define INFV 1.0e9f

// focal diff: pos(x) - neg(x) for one logit (approx rcp sigmoid; discrete
// matching output tolerates ULP-level differences)
__device__ __forceinline__ float focal_diff(float x) {
    float e   = __expf(-x);
    float p   = __builtin_amdgcn_rcpf(1.0f + e);
    float omp = 1.0f - p;
    float pos = 0.25f * omp * omp * (-__logf(p   + 1e-8f));
    float neg = 0.75f * p   * p   * (-__logf(omp + 1e-8f));
    return pos - neg;
}

// ---------------------------------------------------------------------------
// Kernel 1: per-image cost matrix, transposed (T x QPAD), class term via
// exact f32 WMMA one-hot GEMM:  D(QxT) = FocalDiff(QxC) @ OneHot(CxT)
// ---------------------------------------------------------------------------
__global__ __launch_bounds__(256) void cost_kernel(
    const float* __restrict__ logits,   // (BS, NQ, NC)
    const float* __restrict__ pboxes,   // (BS, NQ, 4) cxcywh
    const int*   __restrict__ tlabels,  // (BS, NT)
    const float* __restrict__ tboxes,   // (BS, NT, 4) cxcywh
    float* __restrict__ cost)           // (BS, NT, QPAD)
{
    const int b = blockIdx.x;
    __shared__ int   s_id[NT];
    __shared__ float s_tb[NT][4];   // cxcywh
    __shared__ float s_txy[NT][4];  // xyxy
    __shared__ float s_ta[NT];

    const int tid = threadIdx.x;
    if (tid < NT) {
        s_id[tid] = tlabels[b * NT + tid];
        float cx = tboxes[(b * NT + tid) * 4 + 0];
        float cy = tboxes[(b * NT + tid) * 4 + 1];
        float w  = tboxes[(b * NT + tid) * 4 + 2];
        float h  = tboxes[(b * NT + tid) * 4 + 3];
        s_tb[tid][0] = cx; s_tb[tid][1] = cy; s_tb[tid][2] = w; s_tb[tid][3] = h;
        float x1 = cx - 0.5f * w, y1 = cy - 0.5f * h;
        float x2 = cx + 0.5f * w, y2 = cy + 0.5f * h;
        s_txy[tid][0] = x1; s_txy[tid][1] = y1; s_txy[tid][2] = x2; s_txy[tid][3] = y2;
        s_ta[tid] = (x2 - x1) * (y2 - y1);
    }
    __syncthreads();

    const int lane = tid & 31;
    const int wv   = tid >> 5;        // wave id 0..7
    const int half = lane >> 4;       // 0/1
    const int l16  = lane & 15;

    const float* Lb = logits + (size_t)b * NQ * NC;
    // hoisted target ids for B one-hot (t = nt*16 + l16)
    const int id0 = s_id[l16];
    const int id1 = s_id[16 + l16];
    const int id2 = s_id[32 + l16];

    for (int mt = wv; mt < 19; mt += 8) {          // 19 M-tiles cover QPAD
        const int q  = mt * 16 + l16;              // A row for this lane
        const int qs = (q < NQ) ? q : 0;           // clamped safe row
        const bool qok = (q < NQ);
        v8f acc0 = {}, acc1 = {}, acc2 = {};
        for (int k = 0; k < 23; ++k) {             // K-steps of 4 cover C=91
            const int c0 = 4 * k + 2 * half;       // K = vgpr + 2*half
            const int c1 = c0 + 1;
            // branchless: clamped loads + unconditional focal, then select
            const int cs0 = (c0 < NC) ? c0 : 0;
            const int cs1 = (c1 < NC) ? c1 : 0;
            float d0 = focal_diff(Lb[qs * NC + cs0]);
            float d1 = focal_diff(Lb[qs * NC + cs1]);
            v2f a;
            a.x = (qok && (c0 < NC)) ? d0 : 0.0f;
            a.y = (qok && (c1 < NC)) ? d1 : 0.0f;

            v2f b0, b1, b2;                        // one-hot (ids < 91, so no range check)
            b0.x = (id0 == c0) ? 1.0f : 0.0f;  b0.y = (id0 == c1) ? 1.0f : 0.0f;
            b1.x = (id1 == c0) ? 1.0f : 0.0f;  b1.y = (id1 == c1) ? 1.0f : 0.0f;
            b2.x = (id2 == c0) ? 1.0f : 0.0f;  b2.y = (id2 == c1) ? 1.0f : 0.0f;

            acc0 = __builtin_amdgcn_wmma_f32_16x16x4_f32(false, a, false, b0, (short)0, acc0, false, false);
            acc1 = __builtin_amdgcn_wmma_f32_16x16x4_f32(false, a, false, b1, (short)0, acc1, false, false);
            acc2 = __builtin_amdgcn_wmma_f32_16x16x4_f32(false, a, false, b2, (short)0, acc2, false, false);
        }

        // fuse bbox-L1 + GIoU terms and store transposed cost
        #pragma unroll
        for (int nt = 0; nt < 3; ++nt) {
            v8f acc = (nt == 0) ? acc0 : ((nt == 1) ? acc1 : acc2);
            const int t = nt * 16 + l16;
            const float tcx = s_tb[t][0], tcy = s_tb[t][1], tw = s_tb[t][2], th = s_tb[t][3];
            const float tx1 = s_txy[t][0], ty1 = s_txy[t][1], tx2 = s_txy[t][2], ty2 = s_txy[t][3];
            const float ta  = s_ta[t];
            #pragma unroll
            for (int r = 0; r < 8; ++r) {
                const int qq = mt * 16 + r + 8 * half;   // D row: M = r + 8*half
                if (qq < NQ) {
                    const float4 pb = ((const float4*)pboxes)[b * NQ + qq];
                    float cb = fabsf(pb.x - tcx) + fabsf(pb.y - tcy)
                             + fabsf(pb.z - tw)  + fabsf(pb.w - th);
                    float px1 = pb.x - 0.5f * pb.z, py1 = pb.y - 0.5f * pb.w;
                    float px2 = pb.x + 0.5f * pb.z, py2 = pb.y + 0.5f * pb.w;
                    float pa  = (px2 - px1) * (py2 - py1);
                    float ix1 = fmaxf(px1, tx1), iy1 = fmaxf(py1, ty1);
                    float ix2 = fminf(px2, tx2), iy2 = fminf(py2, ty2);
                    float iw  = fmaxf(ix2 - ix1, 0.0f), ih = fmaxf(iy2 - iy1, 0.0f);
                    float inter = iw * ih;
                    float uni   = pa + ta - inter;
                    float iou   = inter / uni;
                    float ex1 = fminf(px1, tx1), ey1 = fminf(py1, ty1);
                    float ex2 = fmaxf(px2, tx2), ey2 = fmaxf(py2, ty2);
                    float ew  = fmaxf(ex2 - ex1, 0.0f), eh = fmaxf(ey2 - ey1, 0.0f);
                    float ea  = ew * eh;
                    float giou = iou - (ea - uni) / ea;
                    cost[((size_t)b * NT + t) * QPAD + qq] =
                        5.0f * cb + 2.0f * acc[r] - 2.0f * giou;
                }
            }
        }
    }
}

// ---------------------------------------------------------------------------
// Kernel 2: per-image Hungarian (shortest augmenting path), 1 wave32 / image.
// Cost tile staged into LDS by the Tensor Data Mover (one tensor_load_to_lds
// DMA of 48*304 f32 = 58368 B), waited with s_wait_tensorcnt.
// ---------------------------------------------------------------------------
__global__ __launch_bounds__(32) void hung_kernel(
    const float* __restrict__ cost,     // (BS, NT, QPAD)
    long long* __restrict__ outq,       // (BS, NT) sorted query indices
    long long* __restrict__ outt)       // (BS, NT) matching target indices
{
    const int b    = blockIdx.x;
    const int lane = threadIdx.x;

    __shared__ float s_cost[NT * QPAD];
    __shared__ float s_u[NT];
    __shared__ float s_v[QPAD];
    __shared__ float s_minv[QPAD];
    __shared__ int   s_way[QPAD];
    __shared__ int   s_used[QPAD];
    __shared__ int   s_p[QPAD];
    __shared__ int   s_r2c[NT];

    // ---- TDM: DMA the whole cost tile global -> LDS (no VGPR traffic) ----
    {
        const unsigned long long ga =
            (unsigned long long)(uintptr_t)(cost + (size_t)b * NT * QPAD);
        const unsigned lds_off = (unsigned)(uintptr_t)&s_cost[0]; // low 32b of a
                                                                  // generic LDS ptr
        const unsigned NELT = NT * QPAD;                          // 14592 f32
        // D# group 0 (ISA 8.3): count=1 | lds_addr | global_addr[56:0] | type=2
        u32x4 g0;
        g0.x = 1u;                                   // count=1, user mode
        g0.y = lds_off;                              // lds_addr (bytes)
        g0.z = (unsigned)(ga & 0xFFFFFFFFull);       // global_addr[31:0]
        g0.w = (unsigned)((ga >> 32) & 0x01FFFFFFull) | 0x80000000u; // [56:32] | type=2
        // D# group 1 (ISA 8.4): data_size=4B; 1 line of NELT elements
        u32x8 g1;
        g1.s0 = 0x00020000u;                         // workgroup_mask=0, data_size=2(4B)
        g1.s1 = (NELT & 0xFFFFu) << 16;              // tensor_dim0[15:0]
        g1.s2 = (NELT >> 16) & 0xFFFFu;              // tensor_dim0[31:16], tensor_dim1=0
        g1.s3 = (NELT & 0xFFFFu) << 16;              // tensor_dim1 hi=0, tile_dim0=NELT
        g1.s4 = 1u;                                  // tile_dim1=1, tile_dim2=0
        g1.s5 = NELT;                                // tensor_dim0_stride[31:0]
        g1.s6 = 0u;                                  // stride0 hi, stride1 lo
        g1.s7 = 0u;                                  // stride1 hi
        asm volatile("tensor_load_to_lds %0, %1"
                     :: "s"(g0), "s"(g1)
                     : "memory");
    }

    // overlap dual-variable init with the DMA
    for (int j = lane; j < QPAD; j += 32) { s_v[j] = 0.0f; s_p[j] = -1; }
    for (int i = lane; i < NT; i += 32) s_u[i] = 0.0f;

    __builtin_amdgcn_s_wait_tensorcnt(0);
    __syncthreads();

    for (int i = 0; i < NT; ++i) {
        for (int j = lane; j < QPAD; j += 32) { s_minv[j] = INFV; s_used[j] = 0; s_way[j] = 0; }
        if (lane == 0) s_p[NQ] = i;
        __syncthreads();

        int j0 = NQ;
        while (true) {
            int i0 = s_p[j0];
            if (i0 < 0) break;
            if (lane == 0) s_used[j0] = 1;
            __syncthreads();

            const float ui0 = s_u[i0];
            const float* crow = s_cost + i0 * QPAD;
            float lmin = INFV;
            int   lj   = 1 << 30;
            for (int j = lane; j < NQ; j += 32) {
                if (!s_used[j]) {
                    float cur = crow[j] - ui0 - s_v[j];
                    float mv  = s_minv[j];
                    if (cur < mv) { mv = cur; s_minv[j] = cur; s_way[j] = j0; }
                    if (mv < lmin) { lmin = mv; lj = j; }
                }
            }
            // wave32 argmin, tie-break to smallest column (matches jnp.argmin)
            for (int off = 16; off > 0; off >>= 1) {
                float om = __shfl_xor(lmin, off, 32);
                int   oj = __shfl_xor(lj,   off, 32);
                if (om < lmin || (om == lmin && oj < lj)) { lmin = om; lj = oj; }
            }
            const float delta = lmin;
            const int   j1    = lj;
            __syncthreads();

            // dual updates; p[j] distinct across used columns -> race-free
            for (int j = lane; j <= NQ; j += 32) {
                if (s_used[j]) { s_u[s_p[j]] += delta; s_v[j] -= delta; }
                else if (j < NQ) s_minv[j] -= delta;
            }
            __syncthreads();
            j0 = j1;
        }
        __syncthreads();
        if (lane == 0) {   // augment along way[] chain
            int jj = j0;
            while (jj != NQ) { int jp = s_way[jj]; s_p[jj] = s_p[jp]; jj = jp; }
        }
        __syncthreads();
    }

    // row2col: target t -> matched query
    for (int j = lane; j < NQ; j += 32) { int r = s_p[j]; if (r >= 0) s_r2c[r] = j; }
    __syncthreads();
    // sort/argsort of 48 distinct values via rank counting
    for (int t = lane; t < NT; t += 32) {
        const int val = s_r2c[t];
        int rank = 0;
        for (int t2 = 0; t2 < NT; ++t2) rank += (s_r2c[t2] < val) ? 1 : 0;
        outq[b * NT + rank] = (long long)val;
        outt[b * NT + rank] = (long long)t;
    }
}

extern "C" void kernel_launch(void* const* d_in, const int* in_sizes, int n_in,
                              void* d_out, int out_size, void* d_ws, size_t ws_size,
                              hipStream_t stream) {
    const float* logits  = (const float*)d_in[0];   // (32,300,91) f32
    const float* pboxes  = (const float*)d_in[1];   // (32,300,4)  f32
    const int*   tlabels = (const int*)  d_in[2];   // (32,48)     i32
    const float* tboxes  = (const float*)d_in[3];   // (32,48,4)   f32

    float* cost = (float*)d_ws;                     // 32*48*304*4 = 1.78 MB
    long long* outq = (long long*)d_out;            // int64 qi (32,48)
    long long* outt = outq + BS * NT;               // int64 ti (32,48)

    cost_kernel<<<dim3(BS), dim3(256), 0, stream>>>(logits, pboxes, tlabels, tboxes, cost);
    hung_kernel<<<dim3(BS), dim3(32),  0, stream>>>(cost, outq, outt);
}